// Net_82188494176671
// MI455X (gfx1250) — compile-verified
//
#include <hip/hip_runtime.h>

// CDNA5 / gfx1250 fp32 WMMA MLP.
// Each wave processes a 32-sample tile (two M=16 WMMA A-tiles that SHARE every
// weight B-chunk load) with V_WMMA_F32_16X16X4_F32. Layer-to-layer D->A
// transpose goes through a private stride-34 (bank-conflict-free) LDS buffer.
// Weights staged once per persistent block; all per-lane loop-invariant
// weights/biases hoisted into registers; ReLU via a single v_max_num_f32
// (inline asm: avoids the IEEE canonicalize+max pair fmaxf/fmed3 legalize to).

typedef __attribute__((ext_vector_type(2))) float v2f;
typedef __attribute__((ext_vector_type(8))) float v8f;

#define HDIM 32
#define NLAYER 8
#define HSTR 34              // padded row stride (floats): conflict-free ds_b64/b32
#define WAVES_PER_BLOCK 4
#define BLOCK 128
#define MAX_BLOCKS 2048
#define TM 32                // samples per wave-iteration (2 x M=16 tiles)

static __device__ __forceinline__ v8f wmma_f32_k4(v2f a, v2f b, v8f c) {
  // (neg_a, A, neg_b, B, c_mod, C, reuse_a, reuse_b)
  return __builtin_amdgcn_wmma_f32_16x16x4_f32(false, a, false, b, (short)0, c,
                                               false, false);
}

static __device__ __forceinline__ float relu1(float v) {
  float r;
  // single-instruction ReLU; reference produces no NaNs so numNaN semantics ok
  asm("v_max_num_f32 %0, 0, %1" : "=v"(r) : "v"(v));
  return r;
}

__global__ __launch_bounds__(BLOCK) void mlp_wmma_f32_kernel(
    const float* __restrict__ x, const float* __restrict__ W0,
    const float* __restrict__ b0, const float* __restrict__ Wh,
    const float* __restrict__ bh, const float* __restrict__ Wout,
    const float* __restrict__ bout, float* __restrict__ out, int N) {
  __shared__ __align__(16) float s_w0[HDIM * 2];
  __shared__ __align__(16) float s_b0[HDIM];
  __shared__ __align__(16) float s_wh[NLAYER * HDIM * HSTR];
  __shared__ __align__(16) float s_bh[NLAYER * HDIM];
  __shared__ __align__(16) float s_wout[HDIM];
  __shared__ float s_bout;
  __shared__ __align__(16) float s_h[WAVES_PER_BLOCK * TM * HSTR];

  const int tid = threadIdx.x;

  // ---- stage all weights/biases into LDS (once per block) ----
  for (int idx = tid; idx < NLAYER * HDIM * HDIM; idx += BLOCK) {
    int row = idx >> 5;  // i*32 + j
    int k = idx & 31;
    s_wh[row * HSTR + k] = Wh[idx];
  }
  if (tid < HDIM * 2) s_w0[tid] = W0[tid];
  if (tid < HDIM) s_b0[tid] = b0[tid];
  for (int idx = tid; idx < NLAYER * HDIM; idx += BLOCK) s_bh[idx] = bh[idx];
  if (tid < HDIM) s_wout[tid] = Wout[tid];
  if (tid == 0) s_bout = bout[0];
  __syncthreads();

  const int lane = tid & 31;
  const int wid = tid >> 5;
  const int ln = lane & 15;  // sample / column index within half-wave
  const int hi = lane >> 4;  // 0 = lanes 0-15, 1 = lanes 16-31
  float* hrow = &s_h[wid * TM * HSTR];  // private 32x32 (stride 34) buffer

  // ---- hoist all loop-invariant per-lane weights/biases into registers ----
  v2f w0sel[2];        // input-layer B chunk (K=2,3 zero-padded in hi lanes)
  float bin[2];        // input-layer bias
  float bhreg[NLAYER][2];
  v2f wv[8];           // Wout slice for this lane-half
#pragma unroll
  for (int jb = 0; jb < 2; ++jb) {
    const int j = jb * 16 + ln;
    float wa = s_w0[j * 2 + 0];
    float wb = s_w0[j * 2 + 1];
    w0sel[jb].x = hi ? 0.f : wa;
    w0sel[jb].y = hi ? 0.f : wb;
    bin[jb] = s_b0[j];
#pragma unroll
    for (int i = 0; i < NLAYER; ++i) bhreg[i][jb] = s_bh[i * HDIM + j];
  }
#pragma unroll
  for (int t = 0; t < 8; ++t) wv[t] = *(const v2f*)&s_wout[hi * 16 + 2 * t];
  const float bout_v = s_bout;

  const float2* xp = (const float2*)x;
  const int tiles = (N + TM - 1) / TM;
  const int gstride = (int)gridDim.x * WAVES_PER_BLOCK;

  for (int tile = (int)blockIdx.x * WAVES_PER_BLOCK + wid; tile < tiles;
       tile += gstride) {
    const int base = tile * TM;

    int nt = tile + gstride;  // prefetch next tile's x (global_prefetch_b8)
    if (nt < tiles) {
      __builtin_prefetch((const void*)&xp[(size_t)nt * TM + ln], 0, 3);
      __builtin_prefetch((const void*)&xp[(size_t)nt * TM + 16 + ln], 0, 3);
    }

    // ---------------- input layer: zero-padded K=4 WMMAs ----------------
    int s0 = base + ln;
    int s1 = base + 16 + ln;
    if (s0 >= N) s0 = N - 1;
    if (s1 >= N) s1 = N - 1;
    float2 x0 = xp[s0];
    float2 x1 = xp[s1];
    v2f a0t, a1t;  // A: lanes 0-15 hold K=0,1; K=2,3 padded 0
    a0t.x = hi ? 0.f : x0.x;
    a0t.y = hi ? 0.f : x0.y;
    a1t.x = hi ? 0.f : x1.x;
    a1t.y = hi ? 0.f : x1.y;
#pragma unroll
    for (int jb = 0; jb < 2; ++jb) {
      const int j = jb * 16 + ln;
      const float bb = bin[jb];
      v8f c0 = {bb, bb, bb, bb, bb, bb, bb, bb};
      v8f c1 = c0;
      c0 = wmma_f32_k4(a0t, w0sel[jb], c0);
      c1 = wmma_f32_k4(a1t, w0sel[jb], c1);
#pragma unroll
      for (int r = 0; r < 8; ++r)
        hrow[(r + 8 * hi) * HSTR + j] = relu1(c0[r]);
#pragma unroll
      for (int r = 0; r < 8; ++r)
        hrow[(16 + r + 8 * hi) * HSTR + j] = relu1(c1[r]);
    }

    // ---------------- 8 hidden layers: 32 WMMAs each, shared B loads ------
#pragma unroll
    for (int i = 0; i < NLAYER; ++i) {
      v2f a0[8], a1[8];  // A[M, K=4cc+2hi .. +1] for both sample tiles
#pragma unroll
      for (int cc = 0; cc < 8; ++cc) {
        a0[cc] = *(const v2f*)&hrow[ln * HSTR + 4 * cc + 2 * hi];
        a1[cc] = *(const v2f*)&hrow[(16 + ln) * HSTR + 4 * cc + 2 * hi];
      }
      const float* wl = &s_wh[i * HDIM * HSTR];
#pragma unroll
      for (int jb = 0; jb < 2; ++jb) {
        const int j = jb * 16 + ln;
        const float bb = bhreg[i][jb];
        v8f c0 = {bb, bb, bb, bb, bb, bb, bb, bb};
        v8f c1 = c0;
#pragma unroll
        for (int cc = 0; cc < 8; ++cc) {
          // B chunk: per lane W[j][4cc+2hi], W[j][4cc+2hi+1] — shared by both tiles
          v2f b = *(const v2f*)&wl[j * HSTR + 4 * cc + 2 * hi];
          c0 = wmma_f32_k4(a0[cc], b, c0);
          c1 = wmma_f32_k4(a1[cc], b, c1);
        }
#pragma unroll
        for (int r = 0; r < 8; ++r)
          hrow[(r + 8 * hi) * HSTR + j] = relu1(c0[r]);
#pragma unroll
        for (int r = 0; r < 8; ++r)
          hrow[(16 + r + 8 * hi) * HSTR + j] = relu1(c1[r]);
      }
    }

    // ---------------- output layer: per-lane half-dot + cross-half reduce --
#pragma unroll
    for (int g = 0; g < 2; ++g) {
      float acc = 0.f;
#pragma unroll
      for (int t = 0; t < 8; ++t) {
        v2f hv = *(const v2f*)&hrow[(g * 16 + ln) * HSTR + hi * 16 + 2 * t];
        acc += hv.x * wv[t].x + hv.y * wv[t].y;
      }
      acc += __shfl_xor(acc, 16, 32);  // sum feature halves (wave32)
      if (hi == 0) {
        int o = base + g * 16 + ln;
        if (o < N) out[o] = acc + bout_v;
      }
    }
  }
}

extern "C" void kernel_launch(void* const* d_in, const int* in_sizes, int n_in,
                              void* d_out, int out_size, void* d_ws,
                              size_t ws_size, hipStream_t stream) {
  const float* x = (const float*)d_in[0];
  const float* W0 = (const float*)d_in[1];
  const float* b0 = (const float*)d_in[2];
  const float* Wh = (const float*)d_in[3];
  const float* bh = (const float*)d_in[4];
  const float* Wout = (const float*)d_in[5];
  const float* bout = (const float*)d_in[6];
  float* out = (float*)d_out;

  const int N = in_sizes[0] / 2;  // x is [N,2]
  long tiles = ((long)N + TM - 1) / TM;
  long want = (tiles + WAVES_PER_BLOCK - 1) / WAVES_PER_BLOCK;
  int blocks = (int)(want < MAX_BLOCKS ? want : MAX_BLOCKS);
  if (blocks < 1) blocks = 1;

  mlp_wmma_f32_kernel<<<blocks, BLOCK, 0, stream>>>(x, W0, b0, Wh, bh, Wout,
                                                    bout, out, N);
}